// V2I_82952998355463
// MI455X (gfx1250) — compile-verified
//
#include <hip/hip_runtime.h>
#include <math.h>

// Problem constants (fixed by the reference setup)
#define BB 64
#define PP 10
#define SS 64
#define KK 32
#define HH 128
#define NN 33      // K+1 nodes
#define TOTAL 2048 // B*K

typedef __attribute__((ext_vector_type(2))) float v2f;
typedef __attribute__((ext_vector_type(8))) float v8f;

// ---------------------------------------------------------------------------
// Kernel 1: build per-(b,n) node arrays.
//   Xctx[(b*33+n), h] : n==0 -> agent_context[b], else ngh_context[clip(start+n-1)]
//   posn[(b*33+n), 2] : n==0 -> 0, else ngh_pos[clip(start+n-1)]
// ---------------------------------------------------------------------------
__global__ void build_nodes_kernel(const float* __restrict__ agent_ctx,
                                   const float* __restrict__ ngh_ctx,
                                   const float* __restrict__ ngh_pos,
                                   const int* __restrict__ sse,
                                   float* __restrict__ Xctx,
                                   float* __restrict__ posn) {
  int row = blockIdx.x;        // 0 .. B*N-1
  int h   = threadIdx.x;       // 0 .. H-1
  int b = row / NN, n = row % NN;
  if (n == 0) {
    Xctx[(size_t)row * HH + h] = agent_ctx[(size_t)b * HH + h];
    if (h < 2) posn[row * 2 + h] = 0.0f;
  } else {
    int idx  = sse[b * 2] + (n - 1);
    int idxc = idx < 0 ? 0 : (idx > TOTAL - 1 ? TOTAL - 1 : idx);
    Xctx[(size_t)row * HH + h] = ngh_ctx[(size_t)idxc * HH + h];
    if (h < 2) posn[row * 2 + h] = ngh_pos[idxc * 2 + h];
  }
}

// ---------------------------------------------------------------------------
// Kernel 2: validity mask vf[b,p,k].
// One block per (b,p), 64 threads (one per lane point s).
// ---------------------------------------------------------------------------
__global__ void valid_kernel(const float* __restrict__ ngh_pos,
                             const float* __restrict__ lanes, // possible_lanes (S, B*P, 2)
                             const int* __restrict__ sse,
                             const unsigned char* __restrict__ valid_neighbor,
                             float* __restrict__ vf) {
  int bp = blockIdx.x;          // b*P + p
  int b  = bp / PP;
  int t  = threadIdx.x;         // 0..63 (= s)
  __shared__ float lx[SS], ly[SS];
  __shared__ int nanflag;
  if (t == 0) nanflag = 0;
  __syncthreads();
  float x = lanes[((size_t)t * (BB * PP) + bp) * 2 + 0];
  float y = lanes[((size_t)t * (BB * PP) + bp) * 2 + 1];
  if (x != x || y != y) atomicOr(&nanflag, 1);
  lx[t] = x; ly[t] = y;
  __syncthreads();
  if (nanflag) { lx[t] = 0.0f; ly[t] = 0.0f; }
  __syncthreads();
  if (t < KK) {
    int idx   = sse[b * 2] + t;
    bool mask = idx < sse[b * 2 + 1];
    int idxc  = idx < 0 ? 0 : (idx > TOTAL - 1 ? TOTAL - 1 : idx);
    float nx = ngh_pos[idxc * 2 + 0], ny = ngh_pos[idxc * 2 + 1];
    float mind2 = 3.0e38f;
    #pragma unroll 8
    for (int s = 0; s < SS; ++s) {
      float dx = nx - lx[s], dy = ny - ly[s];
      mind2 = fminf(mind2, dx * dx + dy * dy);
    }
    bool valid = (sqrtf(mind2) < 5.0f) && mask && (valid_neighbor[b] != 0);
    vf[(size_t)bp * KK + t] = valid ? 1.0f : 0.0f;
  }
}

// ---------------------------------------------------------------------------
// Kernel 3: U/V dual-output GEMM (small: M=2112, Nout=H).
//   a_t = Xctx * Wa^T, j_t = Xctx * Wj^T  (Wa/Wj are column slices of W_msg)
//   U = j_t + pos_t + b_msg ; V = a_t - pos_t
// ---------------------------------------------------------------------------
__global__ void uv_kernel(const float* __restrict__ Xctx,
                          const float* __restrict__ posn,
                          const float* __restrict__ W_msg, // (H, 2H+2) row-major
                          const float* __restrict__ b_msg,
                          float* __restrict__ U,
                          float* __restrict__ V) {
  const int LDW = 2 * HH + 2; // 258
  int tm = blockIdx.x;        // 0..131  (M = B*N = 2112)
  int tn = blockIdx.y;        // 0..7    (Nout = H = 128)
  int lane = threadIdx.x;     // 0..31
  int hi = lane >> 4, l16 = lane & 15;
  const float* xrow  = Xctx + (size_t)(tm * 16 + l16) * HH;
  const float* wrowA = W_msg + (size_t)(tn * 16 + l16) * LDW + 2;
  const float* wrowJ = W_msg + (size_t)(tn * 16 + l16) * LDW + 2 + HH;
  v8f accA = {0.f,0.f,0.f,0.f,0.f,0.f,0.f,0.f};
  v8f accJ = {0.f,0.f,0.f,0.f,0.f,0.f,0.f,0.f};
  for (int k0 = 0; k0 < HH; k0 += 4) {
    int kk = k0 + hi * 2;
    v2f a  = { xrow[kk],  xrow[kk + 1]  };
    v2f bA = { wrowA[kk], wrowA[kk + 1] };
    v2f bJ = { wrowJ[kk], wrowJ[kk + 1] };
    accA = __builtin_amdgcn_wmma_f32_16x16x4_f32(false, a, false, bA, (short)0, accA, false, false);
    accJ = __builtin_amdgcn_wmma_f32_16x16x4_f32(false, a, false, bJ, (short)0, accJ, false, false);
  }
  int h = tn * 16 + l16;
  float wp0 = W_msg[(size_t)h * LDW + 0];
  float wp1 = W_msg[(size_t)h * LDW + 1];
  float bm  = b_msg[h];
  #pragma unroll
  for (int r = 0; r < 8; ++r) {
    int m = tm * 16 + r + 8 * hi;
    float px = posn[m * 2 + 0], py = posn[m * 2 + 1];
    float pt = px * wp0 + py * wp1;
    U[(size_t)m * HH + h] = accJ[r] + pt + bm;
    V[(size_t)m * HH + h] = accA[r] - pt;
  }
}

// ---------------------------------------------------------------------------
// Kernel 4: pooled[b,p,a-1,h] = sum_{j!=a} w[p,j] * relu(u[j,h] + v[a,h])
// One block per (b, a-1); 128 threads (one per h). msg is p-independent.
// ---------------------------------------------------------------------------
__global__ void pooled_kernel(const float* __restrict__ U,
                              const float* __restrict__ V,
                              const float* __restrict__ vf,
                              float* __restrict__ pooled) {
  int b = blockIdx.x;
  int a = blockIdx.y + 1;     // node index 1..K
  int h = threadIdx.x;        // 0..127
  __shared__ float u_s[NN * HH];   // 16.9 KB
  __shared__ float w_s[PP * NN];   // node weights per p
  const float* Ub = U + (size_t)(b * NN) * HH;
  for (int j = 0; j < NN; ++j) u_s[j * HH + h] = Ub[(size_t)j * HH + h];
  for (int i = h; i < PP * NN; i += HH) {
    int p = i / NN, j = i % NN;
    w_s[i] = (j == 0) ? 1.0f : vf[((size_t)b * PP + p) * KK + (j - 1)];
  }
  __syncthreads();
  float va = V[((size_t)b * NN + a) * HH + h];
  float acc[PP];
  #pragma unroll
  for (int p = 0; p < PP; ++p) acc[p] = 0.0f;
  for (int j = 0; j < NN; ++j) {
    float m = u_s[j * HH + h] + va;
    m = m > 0.0f ? m : 0.0f;
    if (j == a) continue;
    #pragma unroll
    for (int p = 0; p < PP; ++p) acc[p] += w_s[p * NN + j] * m;
  }
  #pragma unroll
  for (int p = 0; p < PP; ++p)
    pooled[((((size_t)b * PP + p) * KK) + (a - 1)) * HH + h] = acc[p];
}

// ---------------------------------------------------------------------------
// Kernel 5: generic C = X * W^T + bias via f32 WMMA, 32x32 tile per wave.
// 2x2 grid of 16x16 WMMA tiles: A frags reused across n, B frags across m
// -> 4 b64 loads per 4 wmma per k-step (1.0 loads/wmma).
// M % 32 == 0, Nout % 32 == 0, Kd % 4 == 0.
// ---------------------------------------------------------------------------
__global__ void gemm_xwT_kernel(const float* __restrict__ X, int ldx,
                                const float* __restrict__ W, int ldw,
                                const float* __restrict__ bias,
                                float* __restrict__ C, int ldc,
                                int Kd, int ntn) {
  int tile = blockIdx.x;
  int tm = tile / ntn, tn = tile % ntn;
  int lane = threadIdx.x;
  int hi = lane >> 4, l16 = lane & 15;
  const float* xrow0 = X + (size_t)(tm * 32 + l16)      * ldx;
  const float* xrow1 = X + (size_t)(tm * 32 + 16 + l16) * ldx;
  const float* wrow0 = W + (size_t)(tn * 32 + l16)      * ldw;
  const float* wrow1 = W + (size_t)(tn * 32 + 16 + l16) * ldw;
  v8f acc00 = {0.f,0.f,0.f,0.f,0.f,0.f,0.f,0.f};
  v8f acc01 = {0.f,0.f,0.f,0.f,0.f,0.f,0.f,0.f};
  v8f acc10 = {0.f,0.f,0.f,0.f,0.f,0.f,0.f,0.f};
  v8f acc11 = {0.f,0.f,0.f,0.f,0.f,0.f,0.f,0.f};
  for (int k0 = 0; k0 < Kd; k0 += 4) {
    int kk = k0 + hi * 2;
    v2f a0 = { xrow0[kk], xrow0[kk + 1] };
    v2f a1 = { xrow1[kk], xrow1[kk + 1] };
    v2f b0 = { wrow0[kk], wrow0[kk + 1] };
    v2f b1 = { wrow1[kk], wrow1[kk + 1] };
    acc00 = __builtin_amdgcn_wmma_f32_16x16x4_f32(false, a0, false, b0, (short)0, acc00, false, false);
    acc01 = __builtin_amdgcn_wmma_f32_16x16x4_f32(false, a0, false, b1, (short)0, acc01, false, false);
    acc10 = __builtin_amdgcn_wmma_f32_16x16x4_f32(false, a1, false, b0, (short)0, acc10, false, false);
    acc11 = __builtin_amdgcn_wmma_f32_16x16x4_f32(false, a1, false, b1, (short)0, acc11, false, false);
  }
  int n0 = tn * 32 + l16;
  int n1 = n0 + 16;
  float bv0 = bias[n0];
  float bv1 = bias[n1];
  #pragma unroll
  for (int r = 0; r < 8; ++r) {
    int m0 = tm * 32 + r + 8 * hi;
    int m1 = m0 + 16;
    C[(size_t)m0 * ldc + n0] = acc00[r] + bv0;
    C[(size_t)m0 * ldc + n1] = acc01[r] + bv1;
    C[(size_t)m1 * ldc + n0] = acc10[r] + bv0;
    C[(size_t)m1 * ldc + n1] = acc11[r] + bv1;
  }
}

// ---------------------------------------------------------------------------
// Kernel 6: GRU gates + weighted neighbor sum + lane_recon pass-through.
// One thread per (b,p,h).
// ---------------------------------------------------------------------------
__global__ void gru_out_kernel(const float* __restrict__ Gi,   // (B*P*K, 3H)
                               const float* __restrict__ Ggh,  // (B*N, 3H)
                               const float* __restrict__ Xctx, // (B*N, H)
                               const float* __restrict__ vf,   // (B*P, K)
                               const float* __restrict__ lane_context, // (B*P, H)
                               float* __restrict__ out) {
  int i = blockIdx.x * blockDim.x + threadIdx.x;
  if (i >= BB * PP * HH) return;
  int h  = i % HH;
  int bp = i / HH;
  int b  = bp / PP;
  out[i] = lane_context[i];  // lane_recon
  float accum = 0.0f;
  for (int k = 0; k < KK; ++k) {
    float w = vf[(size_t)bp * KK + k];
    if (w == 0.0f) continue;
    size_t gir = ((size_t)bp * KK + k) * (3 * HH);
    size_t ghr = ((size_t)b * NN + k + 1) * (3 * HH);
    float ir = Gi[gir + h],           iz = Gi[gir + HH + h],  inn = Gi[gir + 2 * HH + h];
    float hr = Ggh[ghr + h],          hz = Ggh[ghr + HH + h], hn  = Ggh[ghr + 2 * HH + h];
    float ctx = Xctx[((size_t)b * NN + k + 1) * HH + h];
    float r = 1.0f / (1.0f + expf(-(ir + hr)));
    float z = 1.0f / (1.0f + expf(-(iz + hz)));
    float nn = tanhf(inn + r * hn);
    float hnew = (1.0f - z) * nn + z * ctx;
    accum += w * hnew;
  }
  out[BB * PP * HH + i] = accum;
}

// ---------------------------------------------------------------------------
extern "C" void kernel_launch(void* const* d_in, const int* in_sizes, int n_in,
                              void* d_out, int out_size, void* d_ws, size_t ws_size,
                              hipStream_t stream) {
  (void)in_sizes; (void)n_in; (void)out_size; (void)ws_size;
  // Inputs per setup_inputs() order:
  const float* agent_ctx  = (const float*)d_in[1];
  const float* ngh_pos    = (const float*)d_in[2];
  const float* ngh_ctx    = (const float*)d_in[3];
  const float* lanes      = (const float*)d_in[4];
  const float* lane_ctx   = (const float*)d_in[5];
  const int*   sse        = (const int*)d_in[7];
  const unsigned char* vn = (const unsigned char*)d_in[8];
  const float* W_msg      = (const float*)d_in[9];
  const float* b_msg      = (const float*)d_in[10];
  const float* W_ih       = (const float*)d_in[11];
  const float* W_hh       = (const float*)d_in[12];
  const float* b_ih       = (const float*)d_in[13];
  const float* b_hh       = (const float*)d_in[14];
  float* out = (float*)d_out;

  // Workspace layout (floats)
  float* ws = (float*)d_ws;
  const size_t ROWS   = (size_t)BB * NN;        // 2112
  const size_t MROWS  = (size_t)BB * PP * KK;   // 20480
  float* vf     = ws;                 size_t off = (size_t)BB * PP * KK;  // 20480
  float* posn   = ws + off;           off += ROWS * 2;                    // 4224
  float* Xctx   = ws + off;           off += ROWS * HH;                   // 270336
  float* U      = ws + off;           off += ROWS * HH;
  float* V      = ws + off;           off += ROWS * HH;
  float* pooled = ws + off;           off += MROWS * HH;                  // 2.62M
  float* Ggh    = ws + off;           off += ROWS * 3 * HH;               // 811K
  float* Gi     = ws + off;           off += MROWS * 3 * HH;              // 7.86M

  // 1) Gather node context / positions
  build_nodes_kernel<<<(int)ROWS, HH, 0, stream>>>(agent_ctx, ngh_ctx, ngh_pos, sse, Xctx, posn);

  // 2) Validity mask vf[b,p,k]
  valid_kernel<<<BB * PP, SS, 0, stream>>>(ngh_pos, lanes, sse, vn, vf);

  // 3) U/V via dual-B WMMA GEMM: M=2112 (132 tiles), Nout=H (8 tiles)
  uv_kernel<<<dim3(ROWS / 16, HH / 16), 32, 0, stream>>>(Xctx, posn, W_msg, b_msg, U, V);

  // 4) pooled[b,p,a,h]
  pooled_kernel<<<dim3(BB, KK), HH, 0, stream>>>(U, V, vf, pooled);

  // 5) gh = Xctx * W_hh^T + b_hh : M=2112 (66 mtiles), Nout=384 (12 ntiles), Kd=128
  gemm_xwT_kernel<<<(int)(ROWS / 32) * (3 * HH / 32), 32, 0, stream>>>(
      Xctx, HH, W_hh, HH, b_hh, Ggh, 3 * HH, HH, 3 * HH / 32);

  // 6) gi = pooled * W_ih^T + b_ih : M=20480 (640 mtiles), Nout=384, Kd=128
  gemm_xwT_kernel<<<(int)(MROWS / 32) * (3 * HH / 32), 32, 0, stream>>>(
      pooled, HH, W_ih, HH, b_ih, Gi, 3 * HH, HH, 3 * HH / 32);

  // 7) GRU + weighted sum + lane_recon
  int nthreads = BB * PP * HH;
  gru_out_kernel<<<(nthreads + 255) / 256, 256, 0, stream>>>(Gi, Ggh, Xctx, vf, lane_ctx, out);
}